// Adjacency_13134009991680
// MI455X (gfx1250) — compile-verified
//
#include <hip/hip_runtime.h>
#include <hip/hip_bf16.h>

// ---- types for WMMA f32 16x16x4 ----
typedef float v2f __attribute__((ext_vector_type(2)));
typedef float v8f __attribute__((ext_vector_type(8)));

#define Pdim 1024
#define Qdim 1024
#define Sdim 128
#define Bdim 8

// =====================================================================
// Kernel 1: pr_part = product @ w1[:S], pe_part = person @ w1[S:]
// One wave (32 lanes) per 16x16 C tile, K-loop of v_wmma_f32_16x16x4_f32.
// Tile index space: 2 matrices x (1024/16) M-tiles x (128/16) N-tiles = 1024 waves.
// =====================================================================
__global__ __launch_bounds__(256) void pp_gemm_wmma(
    const float* __restrict__ prod, const float* __restrict__ pers,
    const float* __restrict__ w1, float* __restrict__ outbuf) {
  const int wavesPerBlock = blockDim.x >> 5;
  const int waveId = blockIdx.x * wavesPerBlock + (threadIdx.x >> 5);
  const int lane = threadIdx.x & 31;

  const int nt    = waveId & 7;          // N tile (128/16)
  const int mt    = (waveId >> 3) & 63;  // M tile (1024/16)
  const int which = waveId >> 9;         // 0 = product, 1 = person

  const float* A = which ? pers : prod;                 // [1024,128] row-major
  const float* W = w1 + which * (Sdim * Sdim);          // [128,128]  row-major
  float*       C = outbuf + which * (Pdim * Sdim);      // [1024,128]

  const int m0   = mt * 16;
  const int n0   = nt * 16;
  const int half = lane >> 4;   // 0: K=k,k+1   1: K=k+2,k+3
  const int l    = lane & 15;   // row (A) / col (B,C)

  const float* arow = A + (size_t)(m0 + l) * Sdim + 2 * half;   // + k
  const float* bcol = W + (size_t)(2 * half) * Sdim + (n0 + l); // + k*128

  v8f acc = {};
#pragma unroll 4
  for (int k = 0; k < Sdim; k += 4) {
    v2f a; a.x = arow[k];            a.y = arow[k + 1];
    v2f b; b.x = bcol[(size_t)k * Sdim]; b.y = bcol[(size_t)(k + 1) * Sdim];
    acc = __builtin_amdgcn_wmma_f32_16x16x4_f32(
        /*neg_a=*/false, a, /*neg_b=*/false, b,
        /*c_mod=*/(short)0, acc, /*reuse_a=*/false, /*reuse_b=*/false);
  }

#pragma unroll
  for (int r = 0; r < 8; ++r) {
    C[(size_t)(m0 + r + 8 * half) * Sdim + (n0 + l)] = acc[r];
  }
}

// =====================================================================
// Kernel 2: adj[p,q] = leaky( softplus(pr[p,:]+pe[q,:]) . w2 ), out = adj * x
// 16x16 (p,q) tile per block; pr/pe tiles + w2 staged in LDS.
// softplus via v_exp_f32 / v_log_f32 (TRANS), then 8 coalesced batch planes.
// =====================================================================
#define SPAD 132  // 128 + 4 pad: stride*4B = 528B (16B aligned, conflict-free)

__global__ __launch_bounds__(256) void adj_softplus_mul(
    const float* __restrict__ x, const float* __restrict__ prpe,
    const float* __restrict__ w2, float* __restrict__ out) {
  __shared__ float s_pr[16 * SPAD];
  __shared__ float s_pe[16 * SPAD];
  __shared__ float s_w2[Sdim];

  const float* pr = prpe;                     // [1024,128]
  const float* pe = prpe + Pdim * Sdim;       // [1024,128]

  const int tid = threadIdx.x;                // 256 threads
  const int qB = blockIdx.x * 16;
  const int pB = blockIdx.y * 16;

  // cooperative LDS fill: 2048 floats each, 8 per thread, coalesced
#pragma unroll
  for (int i = 0; i < 8; ++i) {
    int idx = tid + i * 256;
    int r = idx >> 7, c = idx & 127;
    s_pr[r * SPAD + c] = pr[(size_t)(pB + r) * Sdim + c];
    s_pe[r * SPAD + c] = pe[(size_t)(qB + r) * Sdim + c];
  }
  if (tid < Sdim) s_w2[tid] = w2[tid];
  __syncthreads();

  const int tx = tid & 15;   // q within tile (fast axis -> coalesced)
  const int ty = tid >> 4;   // p within tile
  const float* prow  = &s_pr[ty * SPAD];
  const float* perow = &s_pe[tx * SPAD];

  float acc = 0.f;
#pragma unroll 8
  for (int s = 0; s < Sdim; ++s) {
    float a = prow[s] + perow[s];
    // stable softplus: max(a,0) + log(1 + exp(-|a|))
    float sp = fmaxf(a, 0.f) + __logf(1.f + __expf(-fabsf(a)));
    acc = fmaf(sp, s_w2[s], acc);
  }
  const float adj = (acc >= 0.f) ? acc : 0.1f * acc;

  const size_t base = (size_t)(pB + ty) * Qdim + (qB + tx);
#pragma unroll
  for (int b = 0; b < Bdim; ++b) {
    const size_t off = base + (size_t)b * (Pdim * Qdim);
    out[off] = adj * x[off];
  }
}

// =====================================================================
// launch
// inputs: 0:x [8,1024,1024] f32   1:product [1024,128] f32
//         2:person [1024,128] f32 3:w1 [256,128] f32   4:w2 [128,1] f32
// workspace: pr_part (512KB) + pe_part (512KB) = 1MB
// =====================================================================
extern "C" void kernel_launch(void* const* d_in, const int* in_sizes, int n_in,
                              void* d_out, int out_size, void* d_ws, size_t ws_size,
                              hipStream_t stream) {
  const float* x    = (const float*)d_in[0];
  const float* prod = (const float*)d_in[1];
  const float* pers = (const float*)d_in[2];
  const float* w1   = (const float*)d_in[3];
  const float* w2   = (const float*)d_in[4];
  float* outp = (float*)d_out;
  float* ws   = (float*)d_ws;   // pr_part @ 0, pe_part @ 1024*128

  // Kernel 1: 1024 wave-tiles, 8 waves (256 thr) per block -> 128 blocks
  pp_gemm_wmma<<<dim3(128), dim3(256), 0, stream>>>(prod, pers, w1, ws);

  // Kernel 2: 64 x 64 tiles of 16x16 (p,q)
  adj_softplus_mul<<<dim3(Qdim / 16, Pdim / 16), dim3(256), 0, stream>>>(
      x, ws, w2, outp);
}